// GCNNet_17901423690235
// MI455X (gfx1250) — compile-verified
//
#include <hip/hip_runtime.h>
#include <math.h>

typedef float v2f __attribute__((ext_vector_type(2)));
typedef float v8f __attribute__((ext_vector_type(8)));

#define NFEAT 128
#define NHID  48
#define NCLS  40

// ---------------------------------------------------------------------------
// Degree / normalization kernels
// ---------------------------------------------------------------------------
__global__ void deg_init(float* __restrict__ deg, int n) {
    int i = blockIdx.x * blockDim.x + threadIdx.x;
    if (i < n) deg[i] = 1.0f;  // self-loop contributes 1 to out-degree
}

__global__ void deg_count(const int* __restrict__ row, float* __restrict__ deg, int ne) {
    int e = blockIdx.x * blockDim.x + threadIdx.x;
    if (e < ne) atomicAdd(&deg[row[e]], 1.0f);
}

__global__ void deg_to_dinv(float* __restrict__ deg, int n) {
    int i = blockIdx.x * blockDim.x + threadIdx.x;
    if (i < n) deg[i] = rsqrtf(deg[i]);
}

// ---------------------------------------------------------------------------
// GEMM1: h1[n,48] = x[n,128] @ W1[128,48] + b1   (fp32 WMMA 16x16x4)
// One wave -> 16 rows x 48 cols (3 N-tiles). 8 waves / block -> 128 rows.
// ---------------------------------------------------------------------------
__global__ __launch_bounds__(256) void gemm1_wmma(
    const float* __restrict__ x, const float* __restrict__ W1,
    const float* __restrict__ b1, float* __restrict__ h1, int n)
{
    __shared__ float sW[NFEAT * NHID];   // 24 KB
    __shared__ float sB[NHID];

    int t = threadIdx.x;
    for (int i = t; i < NFEAT * NHID; i += 256) sW[i] = W1[i];
    if (t < NHID) sB[t] = b1[t];
    __syncthreads();

    int wave = t >> 5, lane = t & 31;
    int rowbase = (blockIdx.x * 8 + wave) * 16;
    if (rowbase >= n) return;               // whole-wave uniform exit

    int nl   = lane & 15;
    int koff = (lane >> 4) * 2;             // lanes 0-15: K=kk,kk+1 ; 16-31: K=kk+2,kk+3
    int arow = rowbase + nl;
    if (arow >= n) arow = n - 1;            // clamp (stores are exact-guarded by tiling)
    const float* ap = x + (size_t)arow * NFEAT + koff;

    v8f c0 = {}, c1 = {}, c2 = {};
    for (int kk = 0; kk < NFEAT; kk += 4) {
        v2f a = *(const v2f*)(ap + kk);     // 8B-aligned
        int krow = kk + koff;
        const float* w0 = sW + krow * NHID + nl;
        const float* w1 = w0 + NHID;
        v2f b0 = { w0[0],  w1[0]  };
        v2f b1v = { w0[16], w1[16] };
        v2f b2v = { w0[32], w1[32] };
        c0 = __builtin_amdgcn_wmma_f32_16x16x4_f32(false, a, false, b0,  (short)0, c0, false, false);
        c1 = __builtin_amdgcn_wmma_f32_16x16x4_f32(false, a, false, b1v, (short)0, c1, false, false);
        c2 = __builtin_amdgcn_wmma_f32_16x16x4_f32(false, a, false, b2v, (short)0, c2, false, false);
    }

    int mbase = rowbase + ((lane >> 4) ? 8 : 0);
    float bb0 = sB[nl], bb1 = sB[16 + nl], bb2 = sB[32 + nl];
#pragma unroll
    for (int r = 0; r < 8; ++r) {
        int orow = mbase + r;
        if (orow < n) {
            size_t o = (size_t)orow * NHID;
            h1[o + nl]      = c0[r] + bb0;
            h1[o + 16 + nl] = c1[r] + bb1;
            h1[o + 32 + nl] = c2[r] + bb2;
        }
    }
}

// ---------------------------------------------------------------------------
// GEMM2: h2[n,40] = relu(h1agg[n,48]) @ W2[48,40] + b2   (fp32 WMMA, N padded to 48)
// ---------------------------------------------------------------------------
__global__ __launch_bounds__(256) void gemm2_wmma(
    const float* __restrict__ h1agg, const float* __restrict__ W2,
    const float* __restrict__ b2, float* __restrict__ h2, int n)
{
    __shared__ float sW[NHID * NHID];    // 48x48, cols 40..47 zero-padded
    __shared__ float sB[NHID];

    int t = threadIdx.x;
    for (int i = t; i < NHID * NHID; i += 256) sW[i] = 0.0f;
    if (t < NHID) sB[t] = (t < NCLS) ? b2[t] : 0.0f;
    __syncthreads();
    for (int i = t; i < NHID * NCLS; i += 256) {
        int r = i / NCLS, c = i % NCLS;
        sW[r * NHID + c] = W2[i];
    }
    __syncthreads();

    int wave = t >> 5, lane = t & 31;
    int rowbase = (blockIdx.x * 8 + wave) * 16;
    if (rowbase >= n) return;

    int nl   = lane & 15;
    int koff = (lane >> 4) * 2;
    int arow = rowbase + nl;
    if (arow >= n) arow = n - 1;
    const float* ap = h1agg + (size_t)arow * NHID + koff;

    v8f c0 = {}, c1 = {}, c2 = {};
    for (int kk = 0; kk < NHID; kk += 4) {
        v2f a = *(const v2f*)(ap + kk);
        a.x = fmaxf(a.x, 0.0f);          // fused ReLU
        a.y = fmaxf(a.y, 0.0f);
        int krow = kk + koff;
        const float* w0 = sW + krow * NHID + nl;
        const float* w1 = w0 + NHID;
        v2f b0 = { w0[0],  w1[0]  };
        v2f b1v = { w0[16], w1[16] };
        v2f b2v = { w0[32], w1[32] };
        c0 = __builtin_amdgcn_wmma_f32_16x16x4_f32(false, a, false, b0,  (short)0, c0, false, false);
        c1 = __builtin_amdgcn_wmma_f32_16x16x4_f32(false, a, false, b1v, (short)0, c1, false, false);
        c2 = __builtin_amdgcn_wmma_f32_16x16x4_f32(false, a, false, b2v, (short)0, c2, false, false);
    }

    int mbase = rowbase + ((lane >> 4) ? 8 : 0);
    float bb0 = sB[nl], bb1 = sB[16 + nl], bb2 = sB[32 + nl];
#pragma unroll
    for (int r = 0; r < 8; ++r) {
        int orow = mbase + r;
        if (orow < n) {
            size_t o = (size_t)orow * NCLS;
            h2[o + nl]      = c0[r] + bb0;
            h2[o + 16 + nl] = c1[r] + bb1;
            if (nl < 8) h2[o + 32 + nl] = c2[r] + bb2;   // cols 32..39 only
        }
    }
}

// ---------------------------------------------------------------------------
// Aggregation: dst is *initialized* with the self-loop term (dinv[i]^2 * src[i])
// so no zeroing pass and no atomics for self-loops.
// ---------------------------------------------------------------------------
template <int FEAT>
__global__ void self_loop_init(const float* __restrict__ dinv,
                               const float* __restrict__ src,
                               float* __restrict__ dst, int n)
{
    int idx = blockIdx.x * blockDim.x + threadIdx.x;
    if (idx >= n * FEAT) return;
    int i = idx / FEAT;
    float dv = dinv[i];
    dst[idx] = dv * dv * src[idx];
}

// Edge scatter: TPE threads cooperate per edge, each handles FEAT/TPE floats.
template <int FEAT, int TPE>
__global__ void edge_scatter(const int* __restrict__ row, const int* __restrict__ col,
                             const float* __restrict__ dinv,
                             const float* __restrict__ src, float* __restrict__ dst,
                             int ne)
{
    constexpr int PT = FEAT / TPE;
    int tid = blockIdx.x * blockDim.x + threadIdx.x;
    int e = tid / TPE;
    int l = tid % TPE;
    if (e >= ne) return;
    int r = row[e], c = col[e];
    float nrm = dinv[r] * dinv[c];
    const float* s = src + (size_t)r * FEAT + l * PT;
    float*       d = dst + (size_t)c * FEAT + l * PT;
#pragma unroll
    for (int k = 0; k < PT; ++k) atomicAdd(d + k, nrm * s[k]);
}

// ---------------------------------------------------------------------------
// log_softmax over 40 classes, one wave32 per row, in place.
// ---------------------------------------------------------------------------
__global__ void log_softmax40(float* __restrict__ out, int n)
{
    int wave = threadIdx.x >> 5, lane = threadIdx.x & 31;
    int r = blockIdx.x * 8 + wave;
    if (r >= n) return;
    float* p = out + (size_t)r * NCLS;
    float v1 = p[lane];
    float v2 = (lane < 8) ? p[32 + lane] : -3.402823e38f;
    float m = fmaxf(v1, v2);
#pragma unroll
    for (int off = 16; off; off >>= 1) m = fmaxf(m, __shfl_xor(m, off, 32));
    float s = expf(v1 - m) + ((lane < 8) ? expf(v2 - m) : 0.0f);
#pragma unroll
    for (int off = 16; off; off >>= 1) s += __shfl_xor(s, off, 32);
    float lse = logf(s);
    p[lane] = v1 - m - lse;
    if (lane < 8) p[32 + lane] = v2 - m - lse;
}

// ---------------------------------------------------------------------------
// Launch
// ---------------------------------------------------------------------------
extern "C" void kernel_launch(void* const* d_in, const int* in_sizes, int n_in,
                              void* d_out, int out_size, void* d_ws, size_t ws_size,
                              hipStream_t stream)
{
    const float* x  = (const float*)d_in[0];
    const int*   ei = (const int*)d_in[1];
    const float* W1 = (const float*)d_in[2];
    const float* b1 = (const float*)d_in[3];
    const float* W2 = (const float*)d_in[4];
    const float* b2 = (const float*)d_in[5];
    float* out = (float*)d_out;

    int n  = in_sizes[0] / NFEAT;    // 100000
    int ne = in_sizes[1] / 2;        // 1600000
    const int* row = ei;
    const int* col = ei + ne;

    float* dinv  = (float*)d_ws;                       // n floats
    float* h1    = dinv + 102400;                      // n*48
    float* h1agg = h1 + (size_t)n * NHID;              // n*48
    float* h2pre = h1;                                 // reuse h1 region

    dim3 blk(256);

    // 1) degrees -> dinv
    deg_init<<<dim3((n + 255) / 256), blk, 0, stream>>>(dinv, n);
    deg_count<<<dim3((ne + 255) / 256), blk, 0, stream>>>(row, dinv, ne);
    deg_to_dinv<<<dim3((n + 255) / 256), blk, 0, stream>>>(dinv, n);

    // 2) layer 1 GEMM (WMMA fp32)
    gemm1_wmma<<<dim3((n + 127) / 128), blk, 0, stream>>>(x, W1, b1, h1, n);

    // 3) layer 1 aggregation (self-loop init + edge scatter)
    self_loop_init<NHID><<<dim3((n * NHID + 255) / 256), blk, 0, stream>>>(dinv, h1, h1agg, n);
    edge_scatter<NHID, 16><<<dim3((ne * 16 + 255) / 256), blk, 0, stream>>>(row, col, dinv, h1, h1agg, ne);

    // 4) layer 2 GEMM (fused ReLU on A operand)
    gemm2_wmma<<<dim3((n + 127) / 128), blk, 0, stream>>>(h1agg, W2, b2, h2pre, n);

    // 5) layer 2 aggregation directly into d_out
    self_loop_init<NCLS><<<dim3((n * NCLS + 255) / 256), blk, 0, stream>>>(dinv, h2pre, out, n);
    edge_scatter<NCLS, 8><<<dim3((ne * 8 + 255) / 256), blk, 0, stream>>>(row, col, dinv, h2pre, out, ne);

    // 6) log-softmax in place
    log_softmax40<<<dim3((n + 7) / 8), blk, 0, stream>>>(out, n);
}